// RDF_79379585565599
// MI455X (gfx1250) — compile-verified
//
#include <hip/hip_runtime.h>

// RDF pair-histogram on CDNA5 (gfx1250, wave32) using V_WMMA_F32_16X16X4_F32
// for the cdist identity d^2 = |a|^2 - 2 a.b + |b|^2.
// v3: coordinates pre-scaled by 20 (=1/DELTA) so WMMA yields (20*dist)^2 and
// bin = floor(sqrt(.)); diagonal self-pairs (bin 0 by construction) corrected
// in the finalize kernel instead of masked in the hot loop; dummy-bin routing
// via a single v_min_u32.

typedef __attribute__((ext_vector_type(2))) float v2f;
typedef __attribute__((ext_vector_type(8))) float v8f;

#define TOTAL_BINS 200
#define HIST_PITCH 256       // per-wave histogram pitch; bin 200 = trash bin
#define DUMMY_BIN  200u
#define N0 4096
#define N1 8192
// M-tile ranges: pair0 (0,0): tiles [0,256), pair1 (cross): [256,512), pair2 (1,1): [512,1024)
#define MT0 256
#define MT1 256

__global__ void rdf_zero_ws(unsigned int* __restrict__ ws) {
    int i = blockIdx.x * blockDim.x + threadIdx.x;
    if (i < 3 * TOTAL_BINS) ws[i] = 0u;
}

__global__ __launch_bounds__(256) void rdf_pair_kernel(
    const float* __restrict__ pos0, const float* __restrict__ pos1,
    const float* __restrict__ real_size, unsigned int* __restrict__ ws)
{
    __shared__ unsigned int hist[8][HIST_PITCH];   // per-wave privatized histograms
    const int tid  = threadIdx.x;
    const int wave = tid >> 5;     // 8 wave32s per block
    const int lane = tid & 31;
    const int lmod = lane & 15;
    const int hi   = lane >> 4;    // 0: K=0,1 half ; 1: K=2,3 half

    for (int b = tid; b < 8 * HIST_PITCH; b += 256) ((unsigned int*)hist)[b] = 0u;
    __syncthreads();

    // global wave id -> (pair, m_tile). All 8 waves of a block share one pair
    // because pair boundaries (256, 512) are multiples of 8.
    const int gw = blockIdx.x * 8 + wave;
    int pair, m_tile, nb;
    const float* __restrict__ pa;
    const float* __restrict__ pb;
    if (gw < MT0)            { pair = 0; m_tile = gw;               pa = pos0; pb = pos0; nb = N0; }
    else if (gw < MT0 + MT1) { pair = 1; m_tile = gw - MT0;         pa = pos0; pb = pos1; nb = N1; }
    else                     { pair = 2; m_tile = gw - (MT0 + MT1); pa = pos1; pb = pos1; nb = N1; }
    const int n_tiles = nb >> 4;

    // Pre-scale by 1/DELTA = 20 so the WMMA computes (20*dist)^2 directly and
    // binning is a bare sqrt+trunc (no per-element rescale).
    const float rs0 = real_size[0] * 20.0f;
    const float rs1 = real_size[1] * 20.0f;
    const float rs2 = real_size[2] * 20.0f;

    // ---- A operand (reused across the whole N loop) ----
    // A row m = [-2ax, -2ay, -2az, |a|^2] ; 16x4 f32 layout:
    //   lanes 0-15:  V0=K0, V1=K1 ; lanes 16-31: V0=K2, V1=K3
    const int mrow = (m_tile << 4) + lmod;
    const float ax = pa[3 * mrow + 0] * rs0;
    const float ay = pa[3 * mrow + 1] * rs1;
    const float az = pa[3 * mrow + 2] * rs2;
    const float a2 = ax * ax + ay * ay + az * az;
    v2f A;
    A[0] = hi ? (-2.0f * az) : (-2.0f * ax);   // selects, not branches: keep EXEC all-ones
    A[1] = hi ? a2           : (-2.0f * ay);

    // Split the N-tile loop across gridDim.y chunks for occupancy.
    const int chunks  = (int)gridDim.y;
    const int per     = (n_tiles + chunks - 1) / chunks;
    const int n_begin = (int)blockIdx.y * per;
    int n_end = n_begin + per;
    if (n_end > n_tiles) n_end = n_tiles;

    if (n_begin < n_end) {
        // Software pipeline: preload first B point, fetch next while computing.
        int ncol = (n_begin << 4) + lmod;
        float px = pb[3 * ncol + 0];
        float py = pb[3 * ncol + 1];
        float pz = pb[3 * ncol + 2];

        for (int nt = n_begin; nt < n_end; ++nt) {
            // issue next tile's load early (clamped; overlaps with WMMA+histogram)
            const int ntn   = (nt + 1 < n_end) ? (nt + 1) : nt;
            const int ncoln = (ntn << 4) + lmod;
            const float nx = pb[3 * ncoln + 0];
            const float ny = pb[3 * ncoln + 1];
            const float nz = pb[3 * ncoln + 2];

            // ---- B operand: col n = [bx, by, bz, 1] ----
            const float bx = px * rs0;
            const float by = py * rs1;
            const float bz = pz * rs2;
            const float b2 = bx * bx + by * by + bz * bz;
            v2f B;
            B[0] = hi ? bz   : bx;
            B[1] = hi ? 1.0f : by;

            v8f c = {};
            // D = A x B : D[m][n] = |a_m|^2 - 2 a_m . b_n   (in (20*dist)^2 units)
            c = __builtin_amdgcn_wmma_f32_16x16x4_f32(
                    /*neg_a=*/false, A, /*neg_b=*/false, B,
                    /*c_mod=*/(short)0, c, /*reuse_a=*/false, /*reuse_b=*/false);

            // C/D layout: this lane covers column N = lmod for rows M = gm_base+r.
            // bin = floor(20*dist) = trunc(v_sqrt_f32(d2')). bin >= 0 always, so
            // out-of-range routing is a single unsigned min into the trash bin.
            // Diagonal self-pairs evaluate to d2'≈0 -> bin 0; corrected in finalize.
            #pragma unroll
            for (int r = 0; r < 8; ++r) {
                const float d2  = c[r] + b2;                      // + |b_n|^2
                const float s   = fmaxf(d2, 0.0f);
                const unsigned int bin = (unsigned int)(int)__builtin_amdgcn_sqrtf(s);
                const unsigned int fb  = bin < DUMMY_BIN ? bin : DUMMY_BIN;  // v_min_u32
                atomicAdd(&hist[wave][fb], 1u);                   // unconditional ds_add
            }

            px = nx; py = ny; pz = nz;
        }
    }

    __syncthreads();
    // Block-level reduction of the 8 per-wave histograms -> global ws (u32 atomics).
    for (int b = tid; b < TOTAL_BINS; b += 256) {
        unsigned int s = 0;
        #pragma unroll
        for (int w = 0; w < 8; ++w) s += hist[w][b];
        if (s) atomicAdd(&ws[pair * TOTAL_BINS + b], s);
    }
}

__global__ void rdf_finalize(const float* __restrict__ count_in,
                             const float* __restrict__ real_size,
                             const unsigned int* __restrict__ ws,
                             float* __restrict__ out)
{
    const int idx = blockIdx.x * blockDim.x + threadIdx.x;
    if (idx >= 2 * 2 * TOTAL_BINS) return;
    const int i = idx / (2 * TOTAL_BINS);
    const int j = (idx / TOTAL_BINS) & 1;
    const int k = idx % TOTAL_BINS;
    // cross histograms (0,1) and (1,0) are identical (same distance multiset,
    // symmetric normalization vol/(N0*N1*slice)) -> both read pair 1.
    const int pair = (i == 0 && j == 0) ? 0 : ((i == 1 && j == 1) ? 2 : 1);
    const float vol = real_size[0] * real_size[1] * real_size[2];
    const float Ni = i ? (float)N1 : (float)N0;
    const float Nj = j ? (float)N1 : (float)N0;
    const float DELTA = 0.05f;
    const float slice = ((float)k + 0.5f) * DELTA * DELTA * 2.0f * 3.14159265358979323846f * 3.0f;
    // Self-pair correction: for i==j the hot loop counted the N diagonal
    // zero-distance pairs into bin 0; the reference excludes them.
    float counts = (float)ws[pair * TOTAL_BINS + k];
    if (k == 0 && pair != 1) counts -= Ni;
    out[idx] = count_in[idx] + counts * vol / (Nj * slice * Ni);
}

extern "C" void kernel_launch(void* const* d_in, const int* in_sizes, int n_in,
                              void* d_out, int out_size, void* d_ws, size_t ws_size,
                              hipStream_t stream) {
    const float* pos0      = (const float*)d_in[0];   // 4096*3 f32
    const float* pos1      = (const float*)d_in[1];   // 8192*3 f32
    const float* count_in  = (const float*)d_in[2];   // 2*2*200 f32
    const float* real_size = (const float*)d_in[3];   // 3 f32
    unsigned int* ws  = (unsigned int*)d_ws;          // 3*200 u32 histogram accumulators
    float*        out = (float*)d_out;                // 800 f32

    rdf_zero_ws<<<dim3(3), 256, 0, stream>>>(ws);
    // 1024 M-tiles total -> 128 blocks of 8 waves, x4 N-chunks for occupancy.
    rdf_pair_kernel<<<dim3(128, 4), 256, 0, stream>>>(pos0, pos1, real_size, ws);
    rdf_finalize<<<dim3(4), 256, 0, stream>>>(count_in, real_size, ws, out);
}